// DCNv2_81295140979514
// MI455X (gfx1250) — compile-verified
//
#include <hip/hip_runtime.h>
#include <hip/hip_bf16.h>
#include <math.h>

// ---------------------------------------------------------------------------
// DCNv2 fused block for gfx1250 (MI455X), wave32 / WMMA, double-buffered LDS.
// B=8, C=128, H=W=80, O=128, K=3 (9 taps), Kdim = C*9 = 1152.
// ---------------------------------------------------------------------------

typedef __attribute__((ext_vector_type(2))) float v2f;
typedef __attribute__((ext_vector_type(8))) float v8f;

#define B_   8
#define C_   128
#define O_   128
#define H_   80
#define W_   80
#define HW_  (H_ * W_)      // 6400
#define KD_  (C_ * 9)       // 1152
#define NPIX (B_ * HW_)     // 51200
#define NOUT (B_ * O_ * HW_)

// ---------------------------------------------------------------------------
// Kernel 0: zero BN accumulators
// ---------------------------------------------------------------------------
__global__ __launch_bounds__(256) void dcn_init_stats(float* gsum, float* gsumsq) {
    int t = threadIdx.x;
    if (t < O_) { gsum[t] = 0.0f; gsumsq[t] = 0.0f; }
}

// ---------------------------------------------------------------------------
// Kernel 1: 3x3 offset conv (27 out channels) -> py, px, sigmoid(mask)
// ---------------------------------------------------------------------------
__global__ __launch_bounds__(256) void dcn_offset_conv(
    const float* __restrict__ x, const float* __restrict__ w_off,
    const float* __restrict__ b_off,
    float* __restrict__ py, float* __restrict__ pxv, float* __restrict__ mval)
{
    int idx = blockIdx.x * 256 + threadIdx.x;
    if (idx >= NPIX) return;
    int b = idx / HW_;
    int p = idx - b * HW_;
    int h = p / W_, w = p - (p / W_) * W_;

    float acc[27];
#pragma unroll
    for (int j = 0; j < 27; ++j) acc[j] = b_off[j];

    const float* xb = x + (size_t)b * C_ * HW_;
    for (int c = 0; c < C_; ++c) {
        float xv[9];
#pragma unroll
        for (int dy = 0; dy < 3; ++dy) {
#pragma unroll
            for (int dx = 0; dx < 3; ++dx) {
                int yy = h - 1 + dy, xx = w - 1 + dx;
                bool ok = (yy >= 0) && (yy < H_) && (xx >= 0) && (xx < W_);
                xv[dy * 3 + dx] = ok ? xb[c * HW_ + yy * W_ + xx] : 0.0f;
            }
        }
        const float* wc = w_off + c * 9;
#pragma unroll 3
        for (int j = 0; j < 27; ++j) {
            const float* wj = wc + j * KD_;  // w_off[j][c][tap]
            float s = acc[j];
#pragma unroll
            for (int t = 0; t < 9; ++t) s = fmaf(xv[t], wj[t], s);
            acc[j] = s;
        }
    }

#pragma unroll
    for (int kk = 0; kk < 9; ++kk) {
        int ky = kk / 3, kx = kk - (kk / 3) * 3;
        float dy = acc[2 * kk];
        float dx = acc[2 * kk + 1];
        float m  = acc[18 + kk];
        size_t o = (size_t)(b * 9 + kk) * HW_ + p;
        py[o]   = dy + (float)(h - 1 + ky);
        pxv[o]  = dx + (float)(w - 1 + kx);
        mval[o] = 1.0f / (1.0f + expf(-m));
    }
}

// ---------------------------------------------------------------------------
// Kernel 2: fused bilinear sampling + WMMA GEMM + bias + BN partial sums.
// Block = 256 threads = 8 wave32; 16 pixels per block, all 128 out channels.
// Double-buffered LDS tile (32ch x 9tap x 16px) overlaps the bilinear gather
// of chunk cc+1 with the v_wmma stream of chunk cc.
// ---------------------------------------------------------------------------

struct DcnShared {
    float  samp[2][288 * 16];   // [buf][(c_local*9+kk)*16 + pix]
    float4 w4[144];             // corner weights (mask & zero-pad folded)
    uint4  o4[144];             // corner BYTE offsets (iy*80+ix)*4
};

// Gather one 32-channel chunk into LDS. `xb` is the block-uniform batch base
// (SGPR pair); corner addresses are unsigned 32-bit BYTE offsets so the
// backend selects saddr + zext(voffset) global loads (no 64-bit VALU chain).
__device__ __forceinline__ void dcn_fill_chunk(
    DcnShared& sh, int buf, int cc, const char* __restrict__ xbb,
    int cl, int px0)
{
    const unsigned chbase = (unsigned)((cc * 32 + cl) * HW_) * 4u;  // bytes
    float* dst = &sh.samp[buf][0];
#pragma unroll 3
    for (int kk = 0; kk < 9; ++kk) {
#pragma unroll
        for (int pp = 0; pp < 2; ++pp) {
            int pix = px0 + pp * 8;
            int rem = kk * 16 + pix;
            float4 wv = sh.w4[rem];
            uint4  ov = sh.o4[rem];
            float c0 = *(const float*)(xbb + (chbase + ov.x));
            float c1 = *(const float*)(xbb + (chbase + ov.y));
            float c2 = *(const float*)(xbb + (chbase + ov.z));
            float c3 = *(const float*)(xbb + (chbase + ov.w));
            float v = wv.x * c0 + wv.y * c1 + wv.z * c2 + wv.w * c3;
            dst[(cl * 9 + kk) * 16 + pix] = v;
        }
    }
}

__global__ __launch_bounds__(256) void dcn_sample_gemm(
    const float* __restrict__ x, const float* __restrict__ w_dcn,
    const float* __restrict__ b_dcn,
    const float* __restrict__ py, const float* __restrict__ pxv,
    const float* __restrict__ mval,
    float* __restrict__ out, float* __restrict__ gsum, float* __restrict__ gsumsq)
{
    __shared__ DcnShared sh;

    const int b    = blockIdx.x / (HW_ / 16);
    const int tile = blockIdx.x - b * (HW_ / 16);
    const int p0   = tile * 16;
    const int tid  = threadIdx.x;

    // ---- Phase A: per (pixel, tap) bilinear descriptors
    if (tid < 144) {
        int kk = tid >> 4, pix = tid & 15;
        size_t oo = (size_t)(b * 9 + kk) * HW_ + (p0 + pix);
        float fy = py[oo], fx = pxv[oo], m = mval[oo];
        float y0 = floorf(fy), x0 = floorf(fx);
        float wy1 = fy - y0, wx1 = fx - x0;
        float wy0 = 1.0f - wy1, wx0 = 1.0f - wx1;
        float vy0 = (y0 >=  0.0f && y0 <= 79.0f) ? 1.0f : 0.0f;
        float vy1 = (y0 >= -1.0f && y0 <= 78.0f) ? 1.0f : 0.0f;
        float vx0 = (x0 >=  0.0f && x0 <= 79.0f) ? 1.0f : 0.0f;
        float vx1 = (x0 >= -1.0f && x0 <= 78.0f) ? 1.0f : 0.0f;
        float4 wv;
        wv.x = wy0 * wx0 * m * vy0 * vx0;
        wv.y = wy0 * wx1 * m * vy0 * vx1;
        wv.z = wy1 * wx0 * m * vy1 * vx0;
        wv.w = wy1 * wx1 * m * vy1 * vx1;
        sh.w4[tid] = wv;
        int iy0 = (int)y0, ix0 = (int)x0;
        int cy0 = iy0 < 0 ? 0 : (iy0 > 79 ? 79 : iy0);
        int cy1 = iy0 + 1 < 0 ? 0 : (iy0 + 1 > 79 ? 79 : iy0 + 1);
        int cx0 = ix0 < 0 ? 0 : (ix0 > 79 ? 79 : ix0);
        int cx1 = ix0 + 1 < 0 ? 0 : (ix0 + 1 > 79 ? 79 : ix0 + 1);
        uint4 ov;
        ov.x = (unsigned)(cy0 * W_ + cx0) * 4u;      // byte offsets
        ov.y = (unsigned)(cy0 * W_ + cx1) * 4u;
        ov.z = (unsigned)(cy1 * W_ + cx0) * 4u;
        ov.w = (unsigned)(cy1 * W_ + cx1) * 4u;
        sh.o4[tid] = ov;
    }

    const int lane = tid & 31;
    const int wv_  = tid >> 5;          // wave -> 16-channel slice of O
    const int o0   = wv_ * 16;
    const int row  = lane & 15;         // A: M row  /  B,D: N (pixel) column
    const int half = lane >> 4;         // K-pair / M-half selector
    const int cl   = tid >> 3;          // gather: input channel within chunk
    const int px0  = tid & 7;           // gather: pixel 0..7 (+8)

    const char* xbb = (const char*)(x + (size_t)b * C_ * HW_);  // uniform base
    v8f acc = {0.f, 0.f, 0.f, 0.f, 0.f, 0.f, 0.f, 0.f};
    const float* wrow = w_dcn + (size_t)(o0 + row) * KD_;  // [O][C*9]

    __syncthreads();                     // params visible
    dcn_fill_chunk(sh, 0, 0, xbb, cl, px0);

    for (int cc = 0; cc < 4; ++cc) {     // 32-channel K chunks
        __syncthreads();                 // chunk cc staged; prev compute done
        if (cc < 3)                      // gather next chunk into other buffer
            dcn_fill_chunk(sh, (cc + 1) & 1, cc + 1, xbb, cl, px0);

        const float* wseg = wrow + cc * 288;
        if (cc < 3)
            __builtin_prefetch(wseg + 288, 0, 1);   // global_prefetch next A seg

        const float* sbuf = &sh.samp[cc & 1][0];
#pragma unroll 4
        for (int ks = 0; ks < 72; ++ks) {
            int kl = ks * 4 + 2 * half;  // lanes 0-15: K0/K1, 16-31: K2/K3
            v2f a;  a.x  = wseg[kl];     // A: 16x4 weights
            a.y  = wseg[kl + 1];
            v2f bb; bb.x = sbuf[kl * 16 + row];      // B: 4x16 sampled tile
            bb.y = sbuf[(kl + 1) * 16 + row];
            acc = __builtin_amdgcn_wmma_f32_16x16x4_f32(
                false, a, false, bb, (short)0, acc, false, false);
        }
    }

    // ---- Epilogue: bias, store pre-BN, BN partial sums (shuffle + atomics)
    const int p = p0 + row;              // pixel for this lane
#pragma unroll
    for (int r = 0; r < 8; ++r) {
        int o = o0 + r + 8 * half;       // D row layout
        float v = acc[r] + b_dcn[o];
        out[((size_t)b * O_ + o) * HW_ + p] = v;
        float t = v, t2 = v * v;
#pragma unroll
        for (int d = 1; d < 16; d <<= 1) {           // reduce 16 pixels
            t  += __shfl_xor(t,  d, 32);
            t2 += __shfl_xor(t2, d, 32);
        }
        if (row == 0) {
            atomicAdd(&gsum[o],   t);
            atomicAdd(&gsumsq[o], t2);
        }
    }
}

// ---------------------------------------------------------------------------
// Kernel 3: BatchNorm (population stats) + SiLU, in place on d_out.
// ---------------------------------------------------------------------------
__global__ __launch_bounds__(256) void dcn_bn_silu(
    float* __restrict__ out, const float* __restrict__ gsum,
    const float* __restrict__ gsumsq, const float* __restrict__ gamma,
    const float* __restrict__ beta)
{
    int idx = blockIdx.x * 256 + threadIdx.x;
    if (idx >= NOUT) return;
    int o = (idx / HW_) & (O_ - 1);
    const float invN = 1.0f / (float)(B_ * HW_);
    float mu  = gsum[o] * invN;
    float var = gsumsq[o] * invN - mu * mu;
    float v = out[idx];
    float y = (v - mu) * rsqrtf(var + 1e-5f) * gamma[o] + beta[o];
    out[idx] = y * (1.0f / (1.0f + expf(-y)));
}

// ---------------------------------------------------------------------------
// Launch
// ---------------------------------------------------------------------------
extern "C" void kernel_launch(void* const* d_in, const int* in_sizes, int n_in,
                              void* d_out, int out_size, void* d_ws, size_t ws_size,
                              hipStream_t stream) {
    (void)in_sizes; (void)n_in; (void)out_size; (void)ws_size;
    const float* x     = (const float*)d_in[0];
    const float* w_dcn = (const float*)d_in[1];
    const float* b_dcn = (const float*)d_in[2];
    const float* w_off = (const float*)d_in[3];
    const float* b_off = (const float*)d_in[4];
    const float* gamma = (const float*)d_in[5];
    const float* beta  = (const float*)d_in[6];
    float* out = (float*)d_out;

    char* ws = (char*)d_ws;
    float* gsum   = (float*)(ws);                    // 128 floats
    float* gsumsq = (float*)(ws + 512);              // 128 floats
    float* py     = (float*)(ws + 1024);             // B*9*HW
    float* pxv    = py  + (size_t)B_ * 9 * HW_;
    float* mval   = pxv + (size_t)B_ * 9 * HW_;      // total ws ~ 5.5 MB

    dcn_init_stats<<<1, 256, 0, stream>>>(gsum, gsumsq);
    dcn_offset_conv<<<(NPIX + 255) / 256, 256, 0, stream>>>(
        x, w_off, b_off, py, pxv, mval);
    dcn_sample_gemm<<<B_ * (HW_ / 16), 256, 0, stream>>>(
        x, w_dcn, b_dcn, py, pxv, mval, out, gsum, gsumsq);
    dcn_bn_silu<<<(NOUT + 255) / 256, 256, 0, stream>>>(
        out, gsum, gsumsq, gamma, beta);
}